// WavetableSynth_83915071029280
// MI455X (gfx1250) — compile-verified
//
#include <hip/hip_runtime.h>

// ---------------- problem constants (from reference) ----------------
#define BATCH 16
#define LEN   48000           // SR*DUR
#define NW    64              // wavetables
#define LWT   512             // wavetable length
#define FSCAL (512.0f / 16000.0f)   // lwt / SR

// ---------------- scan configuration ----------------
#define CHUNK 4096
#define NCH   ((LEN + CHUNK - 1) / CHUNK)     // 12 chunks per batch row
#define SCAN_THREADS 256
#define PER_THREAD   (CHUNK / SCAN_THREADS)   // 16

// ---------------- main kernel configuration ----------------
#define TILES_PER_ROW (LEN / 16)              // 3000
#define NTILES (BATCH * TILES_PER_ROW)        // 48000
#define MAIN_BLOCKS 600
#define MAIN_THREADS 256

typedef float v2f __attribute__((ext_vector_type(2)));
typedef float v8f __attribute__((ext_vector_type(8)));

// Phase 1a: per-(batch, chunk) local inclusive scan of increments.
__global__ void scan_chunks(const float* __restrict__ pitch,
                            float* __restrict__ ws_scan,
                            float* __restrict__ ws_tot) {
    const int b  = blockIdx.y;
    const int ch = blockIdx.x;
    const int t  = threadIdx.x;
    const int base = ch * CHUNK + t * PER_THREAD;

    float v[PER_THREAD];
    float run = 0.f;
#pragma unroll
    for (int i = 0; i < PER_THREAD; ++i) {
        const int l = base + i;
        const float x = (l < LEN) ? pitch[(size_t)b * LEN + l] * FSCAL : 0.f;
        run += x;
        v[i] = run;
    }

    __shared__ float sums[SCAN_THREADS];
    sums[t] = run;
    __syncthreads();
    // Hillis-Steele inclusive scan over thread totals
    for (int off = 1; off < SCAN_THREADS; off <<= 1) {
        const float x = sums[t];
        const float y = (t >= off) ? sums[t - off] : 0.f;
        __syncthreads();
        sums[t] = x + y;
        __syncthreads();
    }
    const float excl = sums[t] - run;
#pragma unroll
    for (int i = 0; i < PER_THREAD; ++i) {
        const int l = base + i;
        if (l < LEN) ws_scan[(size_t)b * LEN + l] = excl + v[i];
    }
    if (t == SCAN_THREADS - 1) ws_tot[b * NCH + ch] = sums[t];
}

// Phase 1b: exclusive scan of chunk totals (tiny).
__global__ void scan_offsets(const float* __restrict__ ws_tot,
                             float* __restrict__ ws_pre) {
    const int t = threadIdx.x;
    if (t < BATCH) {
        float run = 0.f;
        for (int c = 0; c < NCH; ++c) {
            ws_pre[t * NCH + c] = run;
            run += ws_tot[t * NCH + c];
        }
    }
}

// Phase 2: streaming wavetable synthesis with f32 WMMA.
// One wave per 16-sample tile: A(16x64)=attention tile, B(64x16)=interpolated
// wavetable columns (gathered from LDS), out = diag(A x B) * envelope.
__global__ void __launch_bounds__(MAIN_THREADS)
synth_wmma(const float* __restrict__ pitch,
           const float* __restrict__ env,
           const float* __restrict__ att,
           const float* __restrict__ wavetables,
           const float* __restrict__ ws_scan,
           const float* __restrict__ ws_pre,
           float* __restrict__ out) {
    __shared__ float wt[NW * LWT];   // 128 KB, fits in 320 KB WGP LDS
    for (int i = threadIdx.x; i < (NW * LWT) / 4; i += blockDim.x) {
        ((float4*)wt)[i] = ((const float4*)wavetables)[i];
    }
    __syncthreads();

    const int lane = threadIdx.x & 31;
    const int m = lane & 15;     // sample-in-tile for A rows / B cols
    const int h = lane >> 4;     // K-half selector for WMMA layout
    const int wavesPerBlock = blockDim.x >> 5;
    const int waveId = blockIdx.x * wavesPerBlock + (threadIdx.x >> 5);
    const int nWaves = gridDim.x * wavesPerBlock;

    for (int t = waveId; t < NTILES; t += nWaves) {
        const int b  = t / TILES_PER_ROW;
        const int l0 = (t - b * TILES_PER_ROW) << 4;
        const int s  = l0 + m;   // absolute sample position in row (both halves dup)

        // ---- reconstruct phase index (matches reference exactly) ----
        float cum = ws_scan[(size_t)b * LEN + s] + ws_pre[b * NCH + (s >> 12)];
        float idx = cum - pitch[s] * FSCAL;          // minus increment[0, s]
        idx = fmodf(idx, (float)LWT);
        idx = (idx < 0.f) ? idx + (float)LWT : idx;  // jnp.remainder semantics
        idx = (((float)LWT - idx) < 1e-5f) ? 0.f : idx;
        const float low   = floorf(idx);
        const float alpha = idx - low;
        const int il = (int)low;
        const int ih = ((int)ceilf(idx)) & (LWT - 1);

        const float* arow = att + ((size_t)b * LEN + s) * NW;

        // prefetch next tile's attention row for this lane
        const int tn = t + nWaves;
        if (tn < NTILES) {
            const int bn  = tn / TILES_PER_ROW;
            const int l0n = (tn - bn * TILES_PER_ROW) << 4;
            __builtin_prefetch(att + ((size_t)bn * LEN + l0n + m) * NW, 0, 0);
        }

        // ---- accumulate C += A_chunk x B_chunk over K = 64 in steps of 4 ----
        v8f acc = {};
#pragma unroll
        for (int c = 0; c < 16; ++c) {
            const int w0 = 4 * c + 2 * h;
            v2f a;
            a.x = arow[w0];          // A[m][w0]
            a.y = arow[w0 + 1];      // A[m][w0+1]
            const float wl0 = wt[w0 * LWT + il];
            const float wh0 = wt[w0 * LWT + ih];
            const float wl1 = wt[(w0 + 1) * LWT + il];
            const float wh1 = wt[(w0 + 1) * LWT + ih];
            v2f bb;
            bb.x = wl0 + alpha * (wh0 - wl0);   // B[w0][m]
            bb.y = wl1 + alpha * (wh1 - wl1);   // B[w0+1][m]
            acc = __builtin_amdgcn_wmma_f32_16x16x4_f32(
                false, a, false, bb, (short)0, acc, false, false);
        }

        // ---- extract diagonal of C (16x16 f32 C/D layout) ----
        // diag(i): i<8 -> VGPR i, lane i ; i>=8 -> VGPR i-8, lane i+16
        float d = 0.f;
#pragma unroll
        for (int r = 0; r < 8; ++r) {
            d = (lane == r)      ? acc[r] : d;
            d = (lane == 24 + r) ? acc[r] : d;
        }
        if (lane < 8 || lane >= 24) {
            const int smp = (lane < 8) ? lane : (lane - 16);
            const size_t o = (size_t)b * LEN + l0 + smp;
            out[o] = d * env[o];
        }
    }
}

extern "C" void kernel_launch(void* const* d_in, const int* in_sizes, int n_in,
                              void* d_out, int out_size, void* d_ws, size_t ws_size,
                              hipStream_t stream) {
    const float* pitch = (const float*)d_in[0];   // (B, L)
    const float* env   = (const float*)d_in[1];   // (B, L, 1)
    const float* att   = (const float*)d_in[2];   // (B, L, W)
    const float* wtg   = (const float*)d_in[3];   // (W, LWT)
    float* out = (float*)d_out;                   // (B, L, 1)

    float* ws_scan = (float*)d_ws;                        // B*L floats (~3 MB)
    float* ws_tot  = ws_scan + (size_t)BATCH * LEN;       // B*NCH floats
    float* ws_pre  = ws_tot + BATCH * NCH;                // B*NCH floats

    dim3 g1(NCH, BATCH);
    scan_chunks<<<g1, SCAN_THREADS, 0, stream>>>(pitch, ws_scan, ws_tot);
    scan_offsets<<<1, 32, 0, stream>>>(ws_tot, ws_pre);
    synth_wmma<<<MAIN_BLOCKS, MAIN_THREADS, 0, stream>>>(
        pitch, env, att, wtg, ws_scan, ws_pre, out);
}